// ForwardKinematics_61478161875047
// MI455X (gfx1250) — compile-verified
//
#include <hip/hip_runtime.h>
#include <math.h>

// ----------------------------------------------------------------------------
// ForwardKinematics on MI455X (gfx1250)
//
// Roofline: 46.7 MB traffic -> ~2.0 us @ 23.3 TB/s. Compute ~0.75 GFMA fp32 +
// 3.7M sincos -> VALU-bound at a similar few-us scale. Not a GEMM: WMMA's only
// mapping is block-diagonal 4x4 packing at 25% utilization with dependent
// D->A chains (hazard NOPs, ISA 7.12.1) plus per-step cross-lane shuffles --
// strictly worse than the per-lane register affine chain. CDNA5-specific path
// used instead: async global->LDS staging (ASYNCcnt) of the stride-28B AoS
// theta array, so all HBM reads are perfectly coalesced and the strided
// access is absorbed by LDS (stride 7 is coprime with 64 banks -> no
// conflicts). The per-burst +1024B displacement is folded into the async
// instruction's IOFFSET, which the ISA applies to both the LDS destination
// and the global source, so the 7 bursts share one address VGPR pair.
// ----------------------------------------------------------------------------

#define FK_BLOCK 256
#define FK_NA 7

typedef float v4f __attribute__((ext_vector_type(4)));

// One async b32 burst: copies one dword from global address g + IOFFSET to LDS address ldsoff + IOFFSET.
// IOFFSET must be a compile-time immediate (signed 24-bit).
#define FK_ASYNC_LOAD_DWORD(g, ldsoff, ioff)                                   \
  asm volatile("global_load_async_to_lds_b32 %0, %1, off offset:" #ioff        \
               :                                                               \
               : "v"(ldsoff), "v"(g)                                           \
               : "memory")

// T <- T @ R(axis,theta), R = I + s*K + (1-c)*K^2 (exact reference formula)
__device__ __forceinline__ void fk_rotate(float A[3][3], float s, float c,
                                          float x, float y, float z) {
  const float omc = 1.0f - c;
  const float xx = x * x, yy = y * y, zz = z * z;
  const float xy = x * y, xz = x * z, yz = y * z;
  const float R00 = fmaf(-omc, yy + zz, 1.0f);
  const float R11 = fmaf(-omc, xx + zz, 1.0f);
  const float R22 = fmaf(-omc, xx + yy, 1.0f);
  const float R01 = fmaf(omc, xy, -s * z);
  const float R10 = fmaf(omc, xy,  s * z);
  const float R02 = fmaf(omc, xz,  s * y);
  const float R20 = fmaf(omc, xz, -s * y);
  const float R12 = fmaf(omc, yz, -s * x);
  const float R21 = fmaf(omc, yz,  s * x);
#pragma unroll
  for (int r = 0; r < 3; ++r) {
    const float a0 = A[r][0], a1 = A[r][1], a2 = A[r][2];
    A[r][0] = fmaf(a0, R00, fmaf(a1, R10, a2 * R20));
    A[r][1] = fmaf(a0, R01, fmaf(a1, R11, a2 * R21));
    A[r][2] = fmaf(a0, R02, fmaf(a1, R12, a2 * R22));
  }
}

// T <- T @ F, F is a constant 4x4 with bottom row [0,0,0,1] (row-major)
__device__ __forceinline__ void fk_compose_fixed(float A[3][3], float p[3],
                                                 const float* __restrict__ F) {
#pragma unroll
  for (int r = 0; r < 3; ++r) {
    const float a0 = A[r][0], a1 = A[r][1], a2 = A[r][2];
    A[r][0] = fmaf(a0, F[0], fmaf(a1, F[4], a2 * F[8]));
    A[r][1] = fmaf(a0, F[1], fmaf(a1, F[5], a2 * F[9]));
    A[r][2] = fmaf(a0, F[2], fmaf(a1, F[6], a2 * F[10]));
    p[r]    = fmaf(a0, F[3], fmaf(a1, F[7], fmaf(a2, F[11], p[r])));
  }
}

__global__ __launch_bounds__(FK_BLOCK) void fk_kernel(
    const float* __restrict__ jp,   // (B, 7)   joint_positions
    const float* __restrict__ ft,   // (8, 4, 4) fixed_transforms
    const float* __restrict__ ax,   // (7, 3)   joint_axes (unit)
    float* __restrict__ out,        // (B, 4, 4)
    int n) {
  __shared__ float sth[FK_BLOCK * FK_NA];

  const int tid = threadIdx.x;
  const long long base = (long long)blockIdx.x * FK_BLOCK;   // first element
  const float* __restrict__ gsrc = jp + base * FK_NA + tid;  // lane's burst base
  // Low 32 bits of the generic pointer of a __shared__ object equal the
  // workgroup-relative LDS byte offset (flat LDS aperture rule, ISA 10.2),
  // which is exactly what the async instruction's VDST VGPR must hold.
  const unsigned lds0 = (unsigned)(unsigned long long)&sth[tid];

  if (base + FK_BLOCK <= (long long)n) {
    // Uniform fast path: full block, EXEC all ones, no per-lane guards.
    // 7 perfectly coalesced async b32 bursts (1792 contiguous dwords),
    // sharing one address VGPR pair + one LDS-offset VGPR via IOFFSET.
    FK_ASYNC_LOAD_DWORD(gsrc, lds0, 0);
    FK_ASYNC_LOAD_DWORD(gsrc, lds0, 1024);
    FK_ASYNC_LOAD_DWORD(gsrc, lds0, 2048);
    FK_ASYNC_LOAD_DWORD(gsrc, lds0, 3072);
    FK_ASYNC_LOAD_DWORD(gsrc, lds0, 4096);
    FK_ASYNC_LOAD_DWORD(gsrc, lds0, 5120);
    FK_ASYNC_LOAD_DWORD(gsrc, lds0, 6144);
  } else {
    const long long remain = (long long)n * FK_NA - base * FK_NA - tid;
    if (remain > 0 * FK_BLOCK) FK_ASYNC_LOAD_DWORD(gsrc, lds0, 0);
    if (remain > 1 * FK_BLOCK) FK_ASYNC_LOAD_DWORD(gsrc, lds0, 1024);
    if (remain > 2 * FK_BLOCK) FK_ASYNC_LOAD_DWORD(gsrc, lds0, 2048);
    if (remain > 3 * FK_BLOCK) FK_ASYNC_LOAD_DWORD(gsrc, lds0, 3072);
    if (remain > 4 * FK_BLOCK) FK_ASYNC_LOAD_DWORD(gsrc, lds0, 4096);
    if (remain > 5 * FK_BLOCK) FK_ASYNC_LOAD_DWORD(gsrc, lds0, 5120);
    if (remain > 6 * FK_BLOCK) FK_ASYNC_LOAD_DWORD(gsrc, lds0, 6144);
  }
#if __has_builtin(__builtin_amdgcn_s_wait_asynccnt)
  __builtin_amdgcn_s_wait_asynccnt(0);
#else
  asm volatile("s_wait_asynccnt 0x0" ::: "memory");
#endif
  __syncthreads();

  const long long nidx = base + tid;
  if (nidx >= (long long)n) return;

  // ---- Per-element math, fully in registers ------------------------------
  float s[FK_NA], c[FK_NA];
#pragma unroll
  for (int i = 0; i < FK_NA; ++i) {
    sincosf(sth[tid * FK_NA + i], &s[i], &c[i]);
  }

  // T = M0 = F0 @ JT0 : start from F0 (affine), then rotate by R0.
  float A[3][3], p[3];
#pragma unroll
  for (int r = 0; r < 3; ++r) {
    A[r][0] = ft[r * 4 + 0];
    A[r][1] = ft[r * 4 + 1];
    A[r][2] = ft[r * 4 + 2];
    p[r]    = ft[r * 4 + 3];
  }
  fk_rotate(A, s[0], c[0], ax[0], ax[1], ax[2]);

#pragma unroll
  for (int j = 1; j < FK_NA; ++j) {
    fk_compose_fixed(A, p, ft + j * 16);
    fk_rotate(A, s[j], c[j], ax[j * 3 + 0], ax[j * 3 + 1], ax[j * 3 + 2]);
  }
  // Last (passive) link: T <- T @ F7
  fk_compose_fixed(A, p, ft + 7 * 16);

  // ---- 4x nontemporal 128-bit stores (output is write-once) --------------
  v4f* __restrict__ o = (v4f*)(out + nidx * 16);
  v4f r0 = {A[0][0], A[0][1], A[0][2], p[0]};
  v4f r1 = {A[1][0], A[1][1], A[1][2], p[1]};
  v4f r2 = {A[2][0], A[2][1], A[2][2], p[2]};
  v4f r3 = {0.0f, 0.0f, 0.0f, 1.0f};
  __builtin_nontemporal_store(r0, o + 0);
  __builtin_nontemporal_store(r1, o + 1);
  __builtin_nontemporal_store(r2, o + 2);
  __builtin_nontemporal_store(r3, o + 3);
}

extern "C" void kernel_launch(void* const* d_in, const int* in_sizes, int n_in,
                              void* d_out, int out_size, void* d_ws, size_t ws_size,
                              hipStream_t stream) {
  (void)n_in; (void)out_size; (void)d_ws; (void)ws_size;
  const float* jp = (const float*)d_in[0];  // joint_positions (B,7)
  const float* ft = (const float*)d_in[1];  // fixed_transforms (8,4,4)
  const float* ax = (const float*)d_in[2];  // joint_axes (7,3)
  float* out = (float*)d_out;

  const int n = in_sizes[0] / FK_NA;        // B
  const int blocks = (n + FK_BLOCK - 1) / FK_BLOCK;
  fk_kernel<<<blocks, FK_BLOCK, 0, stream>>>(jp, ft, ax, out, n);
}